// GCN_15393162789067
// MI455X (gfx1250) — compile-verified
//
#include <hip/hip_runtime.h>

typedef __attribute__((ext_vector_type(2))) float v2f;
typedef __attribute__((ext_vector_type(8))) float v8f;

#define D 128

// ---------- degree / normalization ----------
__global__ __launch_bounds__(256) void k_deg_init(float* __restrict__ deg, int n) {
  int i = blockIdx.x * 256 + threadIdx.x;
  if (i < n) deg[i] = 1.0f;  // self-loop contributes 1 to every node's degree
}

__global__ __launch_bounds__(256) void k_deg_accum(const int* __restrict__ col,
                                                   float* __restrict__ deg, int e) {
  int i = blockIdx.x * 256 + threadIdx.x;
  if (i < e) atomicAdd(&deg[col[i]], 1.0f);
}

__global__ __launch_bounds__(256) void k_dinv(float* __restrict__ deg, int n) {
  int i = blockIdx.x * 256 + threadIdx.x;
  if (i < n) {
    float d = deg[i];
    deg[i] = d > 0.0f ? rsqrtf(d) : 0.0f;  // in place: deg -> dinv
  }
}

// ---------- GEMM xW = seq @ W (fp32 WMMA), fused self-loop init ----------
// One wave computes one 16x16 tile of xW. Block = 8 waves = the 8 column
// tiles of a 16-row stripe. K-loop: 32 x V_WMMA_F32_16X16X4_F32.
//
// A 16x4 f32 layout: lanes 0-15 = rows M, vgpr{0,1} = K{0,1};
//                    lanes 16-31 = rows M, vgpr{0,1} = K{2,3}.
// B 4x16 layout (symmetric): lanes = cols N, half-wave selects K pair.
// C/D 16x16: vgpr j -> M = j + 8*(lane>=16), N = lane%16.
__global__ __launch_bounds__(256) void k_gemm_selfinit(
    const float* __restrict__ seq, const float* __restrict__ W,
    const float* __restrict__ dinv, float* __restrict__ xW,
    float* __restrict__ out) {
  const int lane  = threadIdx.x & 31;
  const int wave  = threadIdx.x >> 5;       // column tile 0..7
  const int row0  = blockIdx.x * 16;
  const int col0  = wave * 16;
  const int mn    = lane & 15;              // M for A-frag, N for B/C/D-frags
  const int khalf = (lane >> 4) << 1;       // 0 or 2

  const float* arow = seq + (size_t)(row0 + mn) * D;
  v8f acc = {};
#pragma unroll 4
  for (int k = 0; k < D; k += 4) {
    v2f a, b;
    a.x = arow[k + khalf];
    a.y = arow[k + khalf + 1];
    b.x = W[(size_t)(k + khalf) * D + col0 + mn];
    b.y = W[(size_t)(k + khalf + 1) * D + col0 + mn];
    acc = __builtin_amdgcn_wmma_f32_16x16x4_f32(
        /*neg_a=*/false, a, /*neg_b=*/false, b,
        /*c_mod=*/(short)0, acc, /*reuse_a=*/false, /*reuse_b=*/false);
  }

  const int mbase = (lane >> 4) << 3;       // 0 or 8
#pragma unroll
  for (int j = 0; j < 8; ++j) {
    int r = row0 + mbase + j;
    float di = dinv[r];
    size_t off = (size_t)r * D + col0 + mn;
    xW[off]  = acc[j];                      // unscaled, for edge gather
    out[off] = acc[j] * di * di;            // self-loop message (norm = dinv^2)
  }
}

// Scalar tail for row counts not divisible by 16 (not used for N=50000).
__global__ __launch_bounds__(128) void k_tail_rows(
    const float* __restrict__ seq, const float* __restrict__ W,
    const float* __restrict__ dinv, float* __restrict__ xW,
    float* __restrict__ out, int rowStart, int n) {
  int r = rowStart + blockIdx.x;
  if (r >= n) return;
  int c = threadIdx.x;
  float acc = 0.0f;
  for (int k = 0; k < D; ++k) acc += seq[(size_t)r * D + k] * W[(size_t)k * D + c];
  float di = dinv[r];
  xW[(size_t)r * D + c]  = acc;
  out[(size_t)r * D + c] = acc * di * di;
}

// ---------- edge gather + atomic scatter-add ----------
// 32 lanes per edge; each lane moves 4 contiguous floats (float4 gather,
// 4x global_atomic_add_f32 scatter). 512B coalesced per edge each way.
__global__ __launch_bounds__(256) void k_scatter(
    const int* __restrict__ row, const int* __restrict__ col,
    const float* __restrict__ xW, const float* __restrict__ dinv,
    float* __restrict__ out, int e) {
  int idx = blockIdx.x * 256 + threadIdx.x;
  int eid = idx >> 5;
  if (eid >= e) return;
  int lane = idx & 31;
  int s = row[eid];
  int t = col[eid];
  float w = dinv[s] * dinv[t];
  const float4* src = (const float4*)(xW + (size_t)s * D);
  float4 v = src[lane];
  float* dst = out + (size_t)t * D + lane * 4;
  atomicAdd(dst + 0, v.x * w);
  atomicAdd(dst + 1, v.y * w);
  atomicAdd(dst + 2, v.z * w);
  atomicAdd(dst + 3, v.w * w);
}

// ---------- bias + PReLU ----------
__global__ __launch_bounds__(256) void k_finalize(
    float* __restrict__ out, const float* __restrict__ bias,
    const float* __restrict__ prelu, int total) {
  int i = blockIdx.x * 256 + threadIdx.x;
  if (i >= total) return;
  float v = out[i] + bias[i & (D - 1)];
  float pw = prelu[0];
  out[i] = v > 0.0f ? v : pw * v;
}

extern "C" void kernel_launch(void* const* d_in, const int* in_sizes, int n_in,
                              void* d_out, int out_size, void* d_ws, size_t ws_size,
                              hipStream_t stream) {
  const float* seq   = (const float*)d_in[0];
  const int*   eidx  = (const int*)d_in[1];
  const float* W     = (const float*)d_in[2];
  const float* bias  = (const float*)d_in[3];
  const float* prelu = (const float*)d_in[4];

  const int N = in_sizes[0] / D;
  const int E = in_sizes[1] / 2;
  const int* rowp = eidx;        // edge_index[0] = source
  const int* colp = eidx + E;    // edge_index[1] = target

  float* xW  = (float*)d_ws;                 // N*D floats
  float* deg = xW + (size_t)N * D;           // N floats (becomes dinv)
  float* out = (float*)d_out;

  k_deg_init <<<(N + 255) / 256, 256, 0, stream>>>(deg, N);
  k_deg_accum<<<(E + 255) / 256, 256, 0, stream>>>(colp, deg, E);
  k_dinv     <<<(N + 255) / 256, 256, 0, stream>>>(deg, N);

  int nt  = N / 16;
  int rem = N - nt * 16;
  if (nt  > 0) k_gemm_selfinit<<<nt, 256, 0, stream>>>(seq, W, deg, xW, out);
  if (rem > 0) k_tail_rows<<<rem, 128, 0, stream>>>(seq, W, deg, xW, out, nt * 16, N);

  int scatter_blocks = (int)(((size_t)E * 32 + 255) / 256);
  k_scatter <<<scatter_blocks, 256, 0, stream>>>(rowp, colp, xW, deg, out, E);

  int total = N * D;
  k_finalize<<<(total + 255) / 256, 256, 0, stream>>>(out, bias, prelu, total);
}